// GraphAttention_45999099740813
// MI455X (gfx1250) — compile-verified
//
#include <hip/hip_runtime.h>
#include <cstddef>

#define ALPHA   0.2f
#define NEG_INF -9e15f

typedef float v2f __attribute__((ext_vector_type(2)));
typedef float v8f __attribute__((ext_vector_type(8)));

constexpr int B = 2, N = 1024, C = 32, E = 32;

// ---------------------------------------------------------------------------
// Kernel 0: precompute the collapsed per-node scores
//   s1[b][p] = a . leaky( x[b,p] . (W2_top + W2_bot) )
//   s2[b][j] = a . leaky( x[b,(2j)%N] . W2_top + x[b,(2j+1)%N] . W2_bot )
// ---------------------------------------------------------------------------
__global__ __launch_bounds__(256) void gat_scores_kernel(
    const float* __restrict__ x, const float* __restrict__ W2,
    const float* __restrict__ a, float* __restrict__ s1, float* __restrict__ s2)
{
  __shared__ float sW[2 * C * E];   // 64 x 32
  __shared__ float sa[E];
  const int tid = threadIdx.x;
  for (int i = tid; i < 2 * C * E; i += 256) sW[i] = W2[i];
  if (tid < E) sa[tid] = a[tid];
  __syncthreads();

  const int g = blockIdx.x * 256 + tid;   // 0 .. B*N-1
  const int b = g >> 10;
  const int p = g & (N - 1);
  const float* xb = x + (size_t)b * N * C;

  // ---- s1 ----
  {
    float y[E];
#pragma unroll
    for (int e = 0; e < E; ++e) y[e] = 0.f;
    const float* xr = xb + p * C;
    for (int c = 0; c < C; ++c) {
      const float xv = xr[c];
#pragma unroll
      for (int e = 0; e < E; ++e)
        y[e] += xv * (sW[c * E + e] + sW[(c + C) * E + e]);
    }
    float s = 0.f;
#pragma unroll
    for (int e = 0; e < E; ++e) {
      const float v = y[e];
      s += sa[e] * (v > 0.f ? v : ALPHA * v);
    }
    s1[g] = s;
  }
  // ---- s2 ----
  {
    const int p0 = (2 * p) & (N - 1);
    const int p1 = (2 * p + 1) & (N - 1);
    const float* u = xb + p0 * C;
    const float* w = xb + p1 * C;
    float y[E];
#pragma unroll
    for (int e = 0; e < E; ++e) y[e] = 0.f;
    for (int c = 0; c < C; ++c) {
      const float uc = u[c], wc = w[c];
#pragma unroll
      for (int e = 0; e < E; ++e)
        y[e] += uc * sW[c * E + e] + wc * sW[(c + C) * E + e];
    }
    float s = 0.f;
#pragma unroll
    for (int e = 0; e < E; ++e) {
      const float v = y[e];
      s += sa[e] * (v > 0.f ? v : ALPHA * v);
    }
    s2[g] = s;
  }
}

// ---------------------------------------------------------------------------
// Kernel 1: masked softmax + attention @ x via V_WMMA_F32_16X16X4_F32
// One workgroup (8 wave32s) per 16-row tile of one batch.
// ---------------------------------------------------------------------------
__global__ __launch_bounds__(256) void gat_attn_kernel(
    const float* __restrict__ x, const int* __restrict__ adj,
    const float* __restrict__ bias, const float* __restrict__ s1,
    const float* __restrict__ s2, float* __restrict__ out)
{
  __shared__ float Pc[16][128];      // probability tile (16 rows x 128 K)
  __shared__ float red[8][32][8];    // cross-wave accumulator reduction
  __shared__ float sE[N];            // shared exp(s2 - mT) table (high half)
  __shared__ float sm_m[16];         // per-row softmax max   (low half)
  __shared__ float sm_rZ[16];        // per-row 1/sum (0 => all-masked, high)

  const int tid  = threadIdx.x;
  const int wave = tid >> 5;
  const int lane = tid & 31;
  const int blk  = blockIdx.x;
  const int b    = blk >> 6;             // 64 row-tiles per batch
  const int i0   = (blk & 63) << 4;      // tile base row
  const bool lowHalf = (i0 < N / 2);

  const float* xb  = x  + (size_t)b * N * C;
  const float* s1b = s1 + b * N;
  const float* s2b = s2 + b * N;
  const int*  adjb = adj + (size_t)b * N * N;

  // ---- Phase 1: per-row softmax stats ------------------------------------
  if (lowHalf) {
    // Row scores take at most 2 distinct values -> stats = counting.
    for (int pass = 0; pass < 2; ++pass) {
      const int r = wave + 8 * pass;
      const int i = i0 + r;
      const int* arow = adjb + (size_t)i * N;
      const float sA = s1b[2 * i];
      const float sB = s1b[2 * i + 1];
      int cA = 0, cB = 0;
#pragma unroll
      for (int jj = 0; jj < 16; ++jj)            // j < 512
        if (arow[lane + 32 * jj] > 0) ++cA;
#pragma unroll
      for (int jj = 16; jj < 32; ++jj)           // j >= 512
        if (arow[lane + 32 * jj] > 0) ++cB;
      int pack = cA | (cB << 16);
#pragma unroll
      for (int off = 16; off >= 1; off >>= 1) pack += __shfl_xor(pack, off);
      cA = pack & 0xffff;
      cB = pack >> 16;
      float m = cA ? (cB ? fmaxf(sA, sB) : sA) : (cB ? sB : NEG_INF);
      float Z = (float)(N - cA - cB) * expf(NEG_INF - m);  // 0; N if all-masked
      if (cA) Z += (float)cA * expf(sA - m);
      if (cB) Z += (float)cB * expf(sB - m);
      if (lane == 0) { sm_m[r] = m; sm_rZ[r] = 1.0f / Z; }
    }
  } else {
    // All 16 rows share the score vector s2[.]  (softmax is shift-invariant:
    // normalize against the block-wide unmasked max, exp computed ONCE).
    float mT = NEG_INF;
#pragma unroll
    for (int jj = 0; jj < 32; ++jj)
      mT = fmaxf(mT, s2b[lane + 32 * jj]);
#pragma unroll
    for (int off = 16; off >= 1; off >>= 1)
      mT = fmaxf(mT, __shfl_xor(mT, off));
    for (int idx = tid; idx < N; idx += 256)
      sE[idx] = expf(s2b[idx] - mT);             // E <= 1, overflow-safe
    __syncthreads();
    for (int pass = 0; pass < 2; ++pass) {
      const int r = wave + 8 * pass;
      const int* arow = adjb + (size_t)(i0 + r) * N;
      float Z = 0.f;
#pragma unroll
      for (int jj = 0; jj < 32; ++jj) {
        const int j  = lane + 32 * jj;
        const float ev = sE[j];                  // hoisted: unconditional load
        const int   av = arow[j];
        Z += (av > 0) ? ev : 0.f;                // pure v_cndmask, no branch
      }
#pragma unroll
      for (int off = 16; off >= 1; off >>= 1) Z += __shfl_xor(Z, off);
      if (lane == 0) sm_rZ[r] = (Z > 0.f) ? (1.0f / Z) : 0.f;
    }
  }
  __syncthreads();

  // ---- Phase 2: stream K, build P tile in LDS, WMMA accumulate ------------
  const int t    = wave & 1;    // output column tile: cols [16t, 16t+16)
  const int ksub = wave >> 1;   // K sub-range (32 wide) within 128-chunk
  const int rB   = tid >> 4;    // builder row 0..15
  const int l16  = tid & 15;
  const int m16  = lane & 15;
  const int hi   = lane >> 4;   // 0/1 (lane half)

  // Hoisted per-row constants for the builder thread
  const int* browAdj = adjb + (size_t)(i0 + rB) * N;
  const float rZ = sm_rZ[rB];
  float pA = 0.f, pB = 0.f, pM;
  if (lowHalf) {
    const float m_r = sm_m[rB];
    pM = expf(NEG_INF - m_r) * rZ;                   // 0, or 1/N if all-masked
    pA = expf(s1b[2 * (i0 + rB)] - m_r) * rZ;        // unused if all-masked
    pB = expf(s1b[2 * (i0 + rB) + 1] - m_r) * rZ;
  } else {
    pM = (rZ == 0.f) ? (1.0f / (float)N) : 0.f;      // uniform if all-masked
  }

  v8f acc = {};
  for (int chunk = 0; chunk < N / 128; ++chunk) {
    const int kbase = chunk * 128;
    __syncthreads();   // previous chunk's WMMA reads finished
    {
#pragma unroll
      for (int s = 0; s < 8; ++s) {
        const int kk = l16 + s * 16;
        const int j  = kbase + kk;
        const int av = browAdj[j];
        float pv;
        if (lowHalf) {
          pv = (av > 0) ? (j < N / 2 ? pA : pB) : pM;  // register cndmask
        } else {
          const float es = sE[j] * rZ;                 // hoisted LDS load
          pv = (av > 0) ? es : pM;                     // register cndmask
        }
        Pc[rB][kk] = pv;
      }
      // Pull next chunk's adj row segment toward the WGP while WMMAs run
      if (chunk + 1 < N / 128)
        __builtin_prefetch(browAdj + kbase + 128 + l16 * 8, 0, 3);
    }
    __syncthreads();
#pragma unroll
    for (int s = 0; s < 8; ++s) {
      // A: 16x4 f32, A[v][lane] = P[M=lane%16][K = base + v + 2*(lane/16)]
      const int kk0 = ksub * 32 + s * 4 + 2 * hi;
      const v2f Av = *(const v2f*)&Pc[m16][kk0];
      // B: 4x16 f32, B[v][lane] = X[K = base + v + 2*(lane/16)][col]
      const int kg = kbase + kk0;
      const float* xp = xb + (size_t)kg * C + t * 16 + m16;
      v2f Bv;
      Bv.x = xp[0];
      Bv.y = xp[C];
      acc = __builtin_amdgcn_wmma_f32_16x16x4_f32(false, Av, false, Bv,
                                                  (short)0, acc, false, false);
    }
  }

  // ---- Cross-wave K reduction, epilogue (bias + tanh) ---------------------
  __syncthreads();
#pragma unroll
  for (int v = 0; v < 8; ++v) red[wave][lane][v] = acc[v];
  __syncthreads();

  if (wave < 2) {
    const int col  = wave * 16 + m16;
    const float bc = bias[col];
#pragma unroll
    for (int v = 0; v < 8; ++v) {
      const float s = red[wave][lane][v] + red[wave + 2][lane][v]
                    + red[wave + 4][lane][v] + red[wave + 6][lane][v];
      const int M = v + 8 * hi;    // C/D layout: M = v + 8*(lane/16)
      out[((size_t)b * N + (i0 + M)) * C + col] = tanhf(s + bc);
    }
  }
}

// ---------------------------------------------------------------------------
extern "C" void kernel_launch(void* const* d_in, const int* in_sizes, int n_in,
                              void* d_out, int out_size, void* d_ws, size_t ws_size,
                              hipStream_t stream) {
  const float* x    = (const float*)d_in[0];
  const int*   adj  = (const int*)d_in[1];
  // d_in[2] = W1: dead code in the reference, never affects the output
  const float* W2   = (const float*)d_in[3];
  const float* a    = (const float*)d_in[4];
  const float* bias = (const float*)d_in[5];
  float* out = (float*)d_out;

  float* s1 = (float*)d_ws;          // B*N floats
  float* s2 = s1 + B * N;            // B*N floats  (16 KB total scratch)

  gat_scores_kernel<<<(B * N) / 256, 256, 0, stream>>>(x, W2, a, s1, s2);
  gat_attn_kernel<<<B * (N / 16), 256, 0, stream>>>(x, adj, bias, s1, s2, out);
}